// GraphConvBlock_9560597201237
// MI455X (gfx1250) — compile-verified
//
#include <hip/hip_runtime.h>

#define NN 100000
#define NE 1600000
#define CH 64
#define EPSV 1e-5f

typedef float v2f __attribute__((ext_vector_type(2)));
typedef float v8f __attribute__((ext_vector_type(8)));

// ---------------- utility kernels ----------------
__global__ void k_zero(float* __restrict__ p, int n) {
    int i = blockIdx.x * blockDim.x + threadIdx.x;
    if (i < n) p[i] = 0.0f;
}

__global__ void k_deg(const int* __restrict__ dst, float* __restrict__ deg, int e) {
    int i = blockIdx.x * blockDim.x + threadIdx.x;
    if (i < e) atomicAdd(&deg[dst[i]], 1.0f);
}

__global__ void k_dinv(const float* __restrict__ deg, float* __restrict__ dinv, int n) {
    int i = blockIdx.x * blockDim.x + threadIdx.x;
    if (i < n) {
        float d = deg[i];
        dinv[i] = (d > 0.0f) ? rsqrtf(d) : 0.0f;
    }
}

__global__ void k_wedge(const int* __restrict__ src, const int* __restrict__ dst,
                        const float* __restrict__ dinv, float* __restrict__ we, int e) {
    int i = blockIdx.x * blockDim.x + threadIdx.x;
    if (i < e) we[i] = -dinv[src[i]] * dinv[dst[i]];
}

// scatter-add propagation: out[dst] += w_edge * x[src]; 32 lanes per edge, 2 ch/lane
__global__ void k_prop(const int* __restrict__ src, const int* __restrict__ dst,
                       const float* __restrict__ we, const float* __restrict__ xin,
                       float* __restrict__ out) {
    int tid = blockIdx.x * blockDim.x + threadIdx.x;   // NE*32 threads
    int e = tid >> 5;
    int lane = tid & 31;
    int s = src[e], d = dst[e];
    float w = we[e];
    const float2 v = *reinterpret_cast<const float2*>(xin + (long long)s * CH + lane * 2);
    float* o = out + (long long)d * CH + lane * 2;
    atomicAdd(o,     w * v.x);
    atomicAdd(o + 1, w * v.y);
}

// Tx2 = 2*Tx2 - Tx0
__global__ void k_fixup(float* __restrict__ tx2, const float* __restrict__ tx0, int n) {
    int i = blockIdx.x * blockDim.x + threadIdx.x;
    if (i < n) tx2[i] = 2.0f * tx2[i] - tx0[i];
}

// ---------------- fused Cheb GEMM: H = Tx0@W0 + Tx1@W1 + Tx2@W2 + b, ReLU, stats ---
// 1 block = 128 threads = 4 waves; wave w handles output cols [16w,16w+16);
// block handles 16 rows. WMMA f32 16x16x4, accumulate over K=64 in 16 steps x 3 terms.
__global__ __launch_bounds__(128) void k_cheb_gemm(
        const float* __restrict__ tx0, const float* __restrict__ tx1,
        const float* __restrict__ tx2, const float* __restrict__ Wk,
        const float* __restrict__ bias, float* __restrict__ out,
        float* __restrict__ sum, float* __restrict__ sumsq) {
    const int wave  = threadIdx.x >> 5;
    const int lane  = threadIdx.x & 31;
    const int mrow  = lane & 15;     // M (for A) / N (for B) within tile
    const int khalf = lane >> 4;     // 0: K={0,1}, 1: K={2,3} of each 4-step
    const int n0    = wave * 16;
    const int rowbase = blockIdx.x * 16;

    v8f c = {0.f, 0.f, 0.f, 0.f, 0.f, 0.f, 0.f, 0.f};
    const float* txs[3] = {tx0, tx1, tx2};

#pragma unroll
    for (int t = 0; t < 3; ++t) {
        const float* __restrict__ X  = txs[t];
        const float* __restrict__ Wt = Wk + t * CH * CH;   // W[t][cin][cout]
#pragma unroll
        for (int kk = 0; kk < 16; ++kk) {
            const int k0 = kk * 4 + khalf * 2;
            // A: 16x4 fp32 tile of Tx rows
            v2f a = *reinterpret_cast<const v2f*>(X + (long long)(rowbase + mrow) * CH + k0);
            // B: 4x16 fp32 tile of W (K rows x N cols), N along lanes
            v2f b;
            b.x = Wt[(k0 + 0) * CH + n0 + mrow];
            b.y = Wt[(k0 + 1) * CH + n0 + mrow];
            c = __builtin_amdgcn_wmma_f32_16x16x4_f32(
                    false, a, false, b, (short)0, c, false, false);
        }
    }

    const float bb = bias[n0 + mrow];
    float s = 0.0f, s2 = 0.0f;
#pragma unroll
    for (int i = 0; i < 8; ++i) {
        // C/D layout: VGPR i -> M = i (lanes 0-15) or 8+i (lanes 16-31), N = lane&15
        float y = c[i] + bb;
        y = (y > 0.0f) ? y : 0.0f;
        out[(long long)(rowbase + i + 8 * khalf) * CH + n0 + mrow] = y;
        s  += y;
        s2 += y * y;
    }
    atomicAdd(&sum[n0 + mrow], s);
    atomicAdd(&sumsq[n0 + mrow], s2);
}

__global__ void k_finstats(const float* __restrict__ sum, const float* __restrict__ sumsq,
                           float* __restrict__ mean, float* __restrict__ rstd) {
    int ch = threadIdx.x;
    if (ch < CH) {
        float m = sum[ch] * (1.0f / NN);
        float v = sumsq[ch] * (1.0f / NN) - m * m;
        mean[ch] = m;
        rstd[ch] = rsqrtf(v + EPSV);
    }
}

__global__ void k_norm(float* __restrict__ h, const float* __restrict__ mean,
                       const float* __restrict__ rstd, int n) {
    int i = blockIdx.x * blockDim.x + threadIdx.x;
    if (i < n) {
        int ch = i & (CH - 1);
        h[i] = (h[i] - mean[ch]) * rstd[ch];
    }
}

// ---------------- host orchestration ----------------
static void run_layer(const int* src, const int* dstp, const float* we,
                      const float* xin, const float* Wk, const float* bk,
                      float* tx1, float* tx2, float* hout, float* stats,
                      hipStream_t stream) {
    const int FN = NN * CH;                       // 6.4M floats
    k_zero<<<(2 * FN + 255) / 256, 256, 0, stream>>>(tx1, 2 * FN); // tx1+tx2 contiguous
    k_prop<<<(NE * 32) / 256, 256, 0, stream>>>(src, dstp, we, xin, tx1);
    k_prop<<<(NE * 32) / 256, 256, 0, stream>>>(src, dstp, we, tx1, tx2);
    k_fixup<<<(FN + 255) / 256, 256, 0, stream>>>(tx2, xin, FN);
    k_zero<<<1, 256, 0, stream>>>(stats, 128);    // sum[64], sumsq[64]
    k_cheb_gemm<<<NN / 16, 128, 0, stream>>>(xin, tx1, tx2, Wk, bk, hout,
                                             stats, stats + CH);
    k_finstats<<<1, 64, 0, stream>>>(stats, stats + CH, stats + 2 * CH, stats + 3 * CH);
    k_norm<<<(FN + 255) / 256, 256, 0, stream>>>(hout, stats + 2 * CH, stats + 3 * CH, FN);
}

extern "C" void kernel_launch(void* const* d_in, const int* in_sizes, int n_in,
                              void* d_out, int out_size, void* d_ws, size_t ws_size,
                              hipStream_t stream) {
    (void)in_sizes; (void)n_in; (void)out_size; (void)ws_size;
    const float* x   = (const float*)d_in[0];
    const int*   ei  = (const int*)d_in[1];     // [2, NE] row-major
    const float* W1  = (const float*)d_in[2];
    const float* b1  = (const float*)d_in[3];
    const float* W2  = (const float*)d_in[4];
    const float* b2  = (const float*)d_in[5];
    float* out = (float*)d_out;

    const int* src  = ei;
    const int* dstp = ei + NE;

    // workspace layout (floats)
    float* ws   = (float*)d_ws;
    float* deg  = ws;                       // NN
    float* dinv = deg + NN;                 // NN
    float* we   = dinv + NN;                // NE
    float* tx1  = we + NE;                  // NN*CH
    float* tx2  = tx1 + NN * CH;            // NN*CH (contiguous after tx1)
    float* h1   = tx2 + NN * CH;            // NN*CH
    float* stats = h1 + NN * CH;            // 256: sum, sumsq, mean, rstd

    // edge weights: w = -dinv[src]*dinv[dst], in-degree based (shared by both layers)
    k_zero<<<(NN + 255) / 256, 256, 0, stream>>>(deg, NN);
    k_deg<<<(NE + 255) / 256, 256, 0, stream>>>(dstp, deg, NE);
    k_dinv<<<(NN + 255) / 256, 256, 0, stream>>>(deg, dinv, NN);
    k_wedge<<<(NE + 255) / 256, 256, 0, stream>>>(src, dstp, dinv, we, NE);

    // layer 1: x -> h1 (relu + instance-norm)
    run_layer(src, dstp, we, x, W1, b1, tx1, tx2, h1, stats, stream);
    // layer 2: h1 -> out
    run_layer(src, dstp, we, h1, W2, b2, tx1, tx2, out, stats, stream);
}